// InvariantPointAttention_80393197846630
// MI455X (gfx1250) — compile-verified
//
#include <hip/hip_runtime.h>
#include <math.h>

// Invariant Point Attention for MI455X (gfx1250), fp32 throughout.
// All GEMMs use V_WMMA_F32_16X16X4_F32 (wave32, 16x16 tile per wave, K step 4).
// The HBM-bound edge_repr pass (302 MB, read-once) uses double-buffered
// GLOBAL_LOAD_ASYNC_TO_LDS_B128 with non-temporal hint + s_wait_asynccnt.

#define S_   768
#define D_   384
#define E_   128
#define H_   12
#define P_   4
#define HD   4608     // H*D
#define HP3  144      // H*P*3
#define COMB 4752     // HD + HP3

typedef __attribute__((ext_vector_type(2))) float v2f;
typedef __attribute__((ext_vector_type(8))) float v8f;

__device__ __forceinline__ v8f wmma_f32_step(v2f a, v2f b, v8f c) {
  // 8 args: (neg_a, A, neg_b, B, c_mod, C, reuse_a, reuse_b)
  return __builtin_amdgcn_wmma_f32_16x16x4_f32(false, a, false, b, (short)0, c,
                                               false, false);
}

// ---------------------------------------------------------------------------
// Generic C[M,N] = A[M,K] @ W[K,N] + bias  (row-major, all dims %16 / K%4 == 0)
// grid = (N/16, M/16), block = 32 (one wave per 16x16 tile)
// ---------------------------------------------------------------------------
__global__ void gemm_bias(const float* __restrict__ A, const float* __restrict__ W,
                          const float* __restrict__ bias, float* __restrict__ C,
                          int M, int N, int K) {
  const int lane = threadIdx.x;
  const int lm = lane & 15, lh = lane >> 4;
  const int row0 = blockIdx.y * 16;
  const int col0 = blockIdx.x * 16;
  const float* arow = A + (size_t)(row0 + lm) * K;
  v8f acc = {};
  for (int kk = 0; kk < K; kk += 4) {
    v2f a, b;
    a.x = arow[kk + 2 * lh];
    a.y = arow[kk + 2 * lh + 1];
    b.x = W[(size_t)(kk + 2 * lh) * N + col0 + lm];
    b.y = W[(size_t)(kk + 2 * lh + 1) * N + col0 + lm];
    acc = wmma_f32_step(a, b, acc);
  }
  const float bv = bias ? bias[col0 + lm] : 0.0f;
#pragma unroll
  for (int r = 0; r < 8; ++r)
    C[(size_t)(row0 + r + 8 * lh) * N + col0 + lm] = acc[r] + bv;
}

// ---------------------------------------------------------------------------
// Apply per-residue frame: out[s,h,p,l] = sum_k pts[s,h,p,k]*rot[s,k,l] + trans[s,l]
// frames: (S,12) = [trans(3) | rot(9 row-major)]
// ---------------------------------------------------------------------------
__global__ void pt_transform(const float* __restrict__ pts,
                             const float* __restrict__ frames,
                             float* __restrict__ out) {
  int idx = blockIdx.x * blockDim.x + threadIdx.x;
  if (idx >= S_ * H_ * P_) return;
  int s = idx / (H_ * P_);
  int hp = idx % (H_ * P_);
  const float* p = pts + (size_t)s * HP3 + hp * 3;
  const float* f = frames + (size_t)s * 12;
  float* po = out + (size_t)s * HP3 + hp * 3;
#pragma unroll
  for (int l = 0; l < 3; ++l) {
    float acc = f[l];  // translation
#pragma unroll
    for (int kd = 0; kd < 3; ++kd) acc += p[kd] * f[3 + kd * 3 + l];
    po[l] = acc;
  }
}

// ---------------------------------------------------------------------------
// edge bias: logits[h][i][j] = sum_e edge[i,j,e] * we[e,h] + be[h]
// Flattened GEMM: M = S*S rows, K = 128, N = 12 (padded to 16).
// HBM-bound stage: streams 302 MB exactly once. Each wave persists over
// TILES_PER_BLOCK row-tiles; tile t+1 is async-copied into the alternate LDS
// buffer (global_load_async_to_lds_b128, non-temporal) while tile t is
// consumed from LDS by the WMMA pipeline. ASYNCcnt-based synchronization.
// grid = (S*S/16/TILES_PER_BLOCK), block = 32 (single wave -> no barriers)
// ---------------------------------------------------------------------------
#define TILES_PER_BLOCK 8

__global__ void __launch_bounds__(32)
edge_bias_gemm(const float* __restrict__ edge,
               const float* __restrict__ we,
               const float* __restrict__ be,
               float* __restrict__ logits) {
  __shared__ __align__(16) float smem[2][16 * E_];  // 2 x 8 KB tiles
  const int lane = threadIdx.x;
  const int lm = lane & 15, lh = lane >> 4;
  const bool live = (lm < H_);
  const float bv = live ? be[lm] : 0.0f;
  const int tile0 = blockIdx.x * TILES_PER_BLOCK;

  // Hoist B fragments (we: 128x12) into registers once: 32 k-steps x v2f.
  v2f bfrag[32];
#pragma unroll
  for (int s = 0; s < 32; ++s) {
    const int kk = s * 4;
    bfrag[s].x = live ? we[(kk + 2 * lh) * H_ + lm] : 0.0f;
    bfrag[s].y = live ? we[(kk + 2 * lh + 1) * H_ + lm] : 0.0f;
  }

  // Issue async copy of row-tile t into LDS buffer buf (16 rows x 512 B = 8 KB,
  // 16 x b128 per lane). Increments ASYNCcnt by 16 for this wave.
  auto issue_tile = [&](int t, int buf) {
    const char* gbase = (const char*)(edge + (size_t)(tile0 + t) * 16 * E_);
    const uint32_t lbase = (uint32_t)(uintptr_t)(&smem[buf][0]);
#pragma unroll
    for (int it = 0; it < 16; ++it) {
      const uint32_t off = (uint32_t)(lane + it * 32) * 16u;
      const uint64_t gaddr = (uint64_t)(uintptr_t)(gbase + off);
      const uint32_t laddr = lbase + off;
      asm volatile("global_load_async_to_lds_b128 %0, %1, off th:TH_LOAD_NT"
                   :: "v"(laddr), "v"(gaddr)
                   : "memory");
    }
  };

  issue_tile(0, 0);
  for (int t = 0; t < TILES_PER_BLOCK; ++t) {
    const int buf = t & 1;
    if (t + 1 < TILES_PER_BLOCK) {
      issue_tile(t + 1, buf ^ 1);
      // wait for tile t's 16 ops; allow the 16 just-issued to stay in flight
      asm volatile("s_wait_asynccnt 0x10" ::: "memory");
    } else {
      asm volatile("s_wait_asynccnt 0x0" ::: "memory");
    }
    const float* a0 = &smem[buf][0];
    v8f acc = {};
#pragma unroll
    for (int s = 0; s < 32; ++s) {
      const int kk = s * 4;
      v2f a;
      a.x = a0[lm * E_ + kk + 2 * lh];
      a.y = a0[lm * E_ + kk + 2 * lh + 1];
      acc = wmma_f32_step(a, bfrag[s], acc);
    }
    if (live) {
      const int row0 = (tile0 + t) * 16;
#pragma unroll
      for (int r = 0; r < 8; ++r)
        logits[(size_t)lm * (S_ * S_) + row0 + r + 8 * lh] = acc[r] + bv;
    }
  }
}

// ---------------------------------------------------------------------------
// logits[h][i][j] += q_h[i,:]·k_h[j,:]/sqrt(D) - gamma * ||qpg[i,h]-kpg[j,h]||^2
// grid = (S/16, S/16, H), block = 32
// ---------------------------------------------------------------------------
__global__ void attn_logits(const float* __restrict__ q, const float* __restrict__ k,
                            const float* __restrict__ qpg, const float* __restrict__ kpg,
                            const float* __restrict__ gamma, float* __restrict__ logits) {
  const int lane = threadIdx.x;
  const int lm = lane & 15, lh = lane >> 4;
  const int h = blockIdx.z;
  const int i0 = blockIdx.y * 16;
  const int j0 = blockIdx.x * 16;
  const float* qrow = q + (size_t)(i0 + lm) * HD + h * D_;
  const float* krow = k + (size_t)(j0 + lm) * HD + h * D_;
  v8f acc = {};
  for (int kk = 0; kk < D_; kk += 4) {
    v2f a, b;
    a.x = qrow[kk + 2 * lh];
    a.y = qrow[kk + 2 * lh + 1];
    b.x = krow[kk + 2 * lh];
    b.y = krow[kk + 2 * lh + 1];
    acc = wmma_f32_step(a, b, acc);
  }
  const float g = gamma[0];
  const float scale = 1.0f / sqrtf((float)D_);
  // this lane's column j = j0+lm; rows i = i0 + r + 8*lh
  float kp[12];
  const float* kpp = kpg + (size_t)(j0 + lm) * HP3 + h * 12;
#pragma unroll
  for (int t = 0; t < 12; ++t) kp[t] = kpp[t];
  float* lg = logits + (size_t)h * (S_ * S_);
#pragma unroll
  for (int r = 0; r < 8; ++r) {
    const int i = i0 + r + 8 * lh;
    const float* qpp = qpg + (size_t)i * HP3 + h * 12;
    float ssd = 0.0f;
#pragma unroll
    for (int t = 0; t < 12; ++t) {
      float d = qpp[t] - kp[t];
      ssd += d * d;
    }
    const size_t idx = (size_t)i * S_ + (j0 + lm);
    lg[idx] = lg[idx] + acc[r] * scale - g * ssd;
  }
}

// ---------------------------------------------------------------------------
// softmax over j for each (h,i) row of length S. grid = H*S, block = 256.
// ---------------------------------------------------------------------------
__global__ void softmax_rows(float* __restrict__ logits) {
  float* p = logits + (size_t)blockIdx.x * S_;
  __shared__ float red[256];
  const int t = threadIdx.x;
  float m = -1e30f;
  for (int j = t; j < S_; j += 256) m = fmaxf(m, p[j]);
  red[t] = m;
  __syncthreads();
  for (int s = 128; s > 0; s >>= 1) {
    if (t < s) red[t] = fmaxf(red[t], red[t + s]);
    __syncthreads();
  }
  m = red[0];
  __syncthreads();
  float sum = 0.0f;
  for (int j = t; j < S_; j += 256) {
    float e = __expf(p[j] - m);
    p[j] = e;
    sum += e;
  }
  red[t] = sum;
  __syncthreads();
  for (int s = 128; s > 0; s >>= 1) {
    if (t < s) red[t] += red[t + s];
    __syncthreads();
  }
  const float inv = 1.0f / red[0];
  for (int j = t; j < S_; j += 256) p[j] *= inv;
}

// ---------------------------------------------------------------------------
// scalar_out: comb[i, h*D + d] = sum_j attn[h][i][j] * v[j, h*D + d]
// grid = (D/16, S/16, H), block = 32
// ---------------------------------------------------------------------------
__global__ void attn_v_gemm(const float* __restrict__ attn, const float* __restrict__ v,
                            float* __restrict__ comb) {
  const int lane = threadIdx.x;
  const int lm = lane & 15, lh = lane >> 4;
  const int h = blockIdx.z;
  const int i0 = blockIdx.y * 16;
  const int d0 = blockIdx.x * 16;
  const float* arow = attn + (size_t)h * S_ * S_ + (size_t)(i0 + lm) * S_;
  v8f acc = {};
  for (int kk = 0; kk < S_; kk += 4) {
    v2f a, b;
    a.x = arow[kk + 2 * lh];
    a.y = arow[kk + 2 * lh + 1];
    b.x = v[(size_t)(kk + 2 * lh) * HD + h * D_ + d0 + lm];
    b.y = v[(size_t)(kk + 2 * lh + 1) * HD + h * D_ + d0 + lm];
    acc = wmma_f32_step(a, b, acc);
  }
#pragma unroll
  for (int r = 0; r < 8; ++r)
    comb[(size_t)(i0 + r + 8 * lh) * COMB + h * D_ + d0 + lm] = acc[r];
}

// ---------------------------------------------------------------------------
// point_out: comb[i, HD + h*12 + n] = sum_j attn[h][i][j] * vp[j, h*12 + n]
// N = 12 padded to 16. grid = (1, S/16, H), block = 32
// ---------------------------------------------------------------------------
__global__ void attn_vp_gemm(const float* __restrict__ attn, const float* __restrict__ vp,
                             float* __restrict__ comb) {
  const int lane = threadIdx.x;
  const int lm = lane & 15, lh = lane >> 4;
  const int h = blockIdx.z;
  const int i0 = blockIdx.y * 16;
  const float* arow = attn + (size_t)h * S_ * S_ + (size_t)(i0 + lm) * S_;
  const bool live = (lm < 12);
  v8f acc = {};
  for (int kk = 0; kk < S_; kk += 4) {
    v2f a, b;
    a.x = arow[kk + 2 * lh];
    a.y = arow[kk + 2 * lh + 1];
    b.x = live ? vp[(size_t)(kk + 2 * lh) * HP3 + h * 12 + lm] : 0.0f;
    b.y = live ? vp[(size_t)(kk + 2 * lh + 1) * HP3 + h * 12 + lm] : 0.0f;
    acc = wmma_f32_step(a, b, acc);
  }
  if (live) {
#pragma unroll
    for (int r = 0; r < 8; ++r)
      comb[(size_t)(i0 + r + 8 * lh) * COMB + HD + h * 12 + lm] = acc[r];
  }
}

// ---------------------------------------------------------------------------
extern "C" void kernel_launch(void* const* d_in, const int* in_sizes, int n_in,
                              void* d_out, int out_size, void* d_ws, size_t ws_size,
                              hipStream_t stream) {
  (void)in_sizes; (void)n_in; (void)out_size; (void)ws_size;
  const float* node   = (const float*)d_in[0];
  const float* edge   = (const float*)d_in[1];
  const float* frames = (const float*)d_in[2];
  const float* wq  = (const float*)d_in[3];
  const float* bq  = (const float*)d_in[4];
  const float* wk  = (const float*)d_in[5];
  const float* bk  = (const float*)d_in[6];
  const float* wv  = (const float*)d_in[7];
  const float* bv  = (const float*)d_in[8];
  const float* wqp = (const float*)d_in[9];
  const float* bqp = (const float*)d_in[10];
  const float* wkp = (const float*)d_in[11];
  const float* bkp = (const float*)d_in[12];
  const float* wvp = (const float*)d_in[13];
  const float* bvp = (const float*)d_in[14];
  const float* we  = (const float*)d_in[15];
  const float* be  = (const float*)d_in[16];
  const float* wo  = (const float*)d_in[17];
  const float* bo  = (const float*)d_in[18];
  const float* gamma = (const float*)d_in[19];
  float* out = (float*)d_out;

  // workspace layout (floats)
  float* ws = (float*)d_ws;
  float* q      = ws;                       // S*HD
  float* k      = q   + (size_t)S_ * HD;    // S*HD
  float* v      = k   + (size_t)S_ * HD;    // S*HD
  float* qp     = v   + (size_t)S_ * HD;    // S*HP3
  float* kp     = qp  + (size_t)S_ * HP3;   // S*HP3
  float* vp     = kp  + (size_t)S_ * HP3;   // S*HP3
  float* qpg    = vp  + (size_t)S_ * HP3;   // S*HP3
  float* kpg    = qpg + (size_t)S_ * HP3;   // S*HP3
  float* logits = kpg + (size_t)S_ * HP3;   // H*S*S
  float* comb   = logits + (size_t)H_ * S_ * S_;  // S*COMB

  const dim3 wave(32);

  // 1) projections
  gemm_bias<<<dim3(HD / 16, S_ / 16), wave, 0, stream>>>(node, wq, bq, q, S_, HD, D_);
  gemm_bias<<<dim3(HD / 16, S_ / 16), wave, 0, stream>>>(node, wk, bk, k, S_, HD, D_);
  gemm_bias<<<dim3(HD / 16, S_ / 16), wave, 0, stream>>>(node, wv, bv, v, S_, HD, D_);
  gemm_bias<<<dim3(HP3 / 16, S_ / 16), wave, 0, stream>>>(node, wqp, bqp, qp, S_, HP3, D_);
  gemm_bias<<<dim3(HP3 / 16, S_ / 16), wave, 0, stream>>>(node, wkp, bkp, kp, S_, HP3, D_);
  gemm_bias<<<dim3(HP3 / 16, S_ / 16), wave, 0, stream>>>(node, wvp, bvp, vp, S_, HP3, D_);

  // 2) rigid-frame transform of q/k points
  const int npts = S_ * H_ * P_;
  pt_transform<<<(npts + 255) / 256, 256, 0, stream>>>(qp, frames, qpg);
  pt_transform<<<(npts + 255) / 256, 256, 0, stream>>>(kp, frames, kpg);

  // 3) edge bias (async-to-LDS double-buffered HBM stream)
  edge_bias_gemm<<<dim3((S_ * S_) / 16 / TILES_PER_BLOCK), wave, 0, stream>>>(
      edge, we, be, logits);

  // 4) scalar attention + point attention added onto edge bias
  attn_logits<<<dim3(S_ / 16, S_ / 16, H_), wave, 0, stream>>>(q, k, qpg, kpg, gamma, logits);

  // 5) softmax over j
  softmax_rows<<<dim3(H_ * S_), 256, 0, stream>>>(logits);

  // 6) weighted sums
  attn_v_gemm<<<dim3(D_ / 16, S_ / 16, H_), wave, 0, stream>>>(logits, v, comb);
  attn_vp_gemm<<<dim3(1, S_ / 16, H_), wave, 0, stream>>>(logits, vp, comb);

  // 7) output projection
  gemm_bias<<<dim3(D_ / 16, S_ / 16), wave, 0, stream>>>(comb, wo, bo, out, S_, D_, COMB);
}